// Model_70617852280846
// MI455X (gfx1250) — compile-verified
//
#include <hip/hip_runtime.h>
#include <hip/hip_bf16.h>
#include <math.h>

// ---------------------------------------------------------------------------
// DGCNN pipeline for gfx1250 (MI455X):
//   GCN x4 (WMMA f32 GEMM + edge-scatter f32 atomics) -> sortpool (LDS bitonic)
//   -> sim/top-k select -> conv5/conv6 (LDS) -> FC (WMMA f32 GEMM + bias+relu)
// ---------------------------------------------------------------------------

typedef __attribute__((ext_vector_type(2))) float v2f;
typedef __attribute__((ext_vector_type(8))) float v8f;

#define TPB 256

// ---------------- generic helpers ----------------

__global__ void zero_f32_kernel(float* p, long n) {
    long i = (long)blockIdx.x * TPB + threadIdx.x;
    long st = (long)gridDim.x * TPB;
    for (; i < n; i += st) p[i] = 0.0f;
}

__device__ __forceinline__ unsigned encf(float f) {
    unsigned u = __float_as_uint(f);
    return (u & 0x80000000u) ? ~u : (u | 0x80000000u);
}
__device__ __forceinline__ float decf(unsigned u) {
    return (u & 0x80000000u) ? __uint_as_float(u ^ 0x80000000u) : __uint_as_float(~u);
}

__global__ void min_init_kernel(unsigned* p) { *p = 0xFFFFFFFFu; }

__global__ void min_enc_kernel(const float* __restrict__ p, long n, unsigned* out) {
    __shared__ unsigned red[TPB];
    unsigned local = 0xFFFFFFFFu;
    for (long i = (long)blockIdx.x * TPB + threadIdx.x; i < n; i += (long)gridDim.x * TPB)
        local = min(local, encf(p[i]));
    red[threadIdx.x] = local;
    __syncthreads();
    for (int s = TPB / 2; s > 0; s >>= 1) {
        if (threadIdx.x < s) red[threadIdx.x] = min(red[threadIdx.x], red[threadIdx.x + s]);
        __syncthreads();
    }
    if (threadIdx.x == 0) atomicMin(out, red[0]);
}

// ---------------- graph preprocessing ----------------

__global__ void deg_kernel(const int* __restrict__ ei, float* deg, int E) {
    int e = blockIdx.x * TPB + threadIdx.x;
    if (e >= E) return;
    unsafeAtomicAdd(&deg[ei[E + e]], 1.0f);
}

__global__ void dis_kernel(float* dis, int N) {
    int n = blockIdx.x * TPB + threadIdx.x;
    if (n >= N) return;
    dis[n] = rsqrtf(dis[n] + 1.0f);   // in-place: deg -> dis
}

__global__ void wedge_kernel(const int* __restrict__ ei, const float* __restrict__ dis,
                             float* wedge, int E) {
    int e = blockIdx.x * TPB + threadIdx.x;
    if (e >= E) return;
    wedge[e] = dis[ei[e]] * dis[ei[E + e]];
}

__global__ void count_kernel(const int* __restrict__ batch, int* nn, int N) {
    int n = blockIdx.x * TPB + threadIdx.x;
    if (n >= N) return;
    atomicAdd(&nn[batch[n]], 1);
}

__global__ void offsets_kernel(const int* __restrict__ nn, int* offs, int B) {
    if (blockIdx.x == 0 && threadIdx.x == 0) {
        int a = 0;
        for (int b = 0; b < B; ++b) { offs[b] = a; a += nn[b]; }
        offs[B] = a;
    }
}

// ---------------- WMMA f32 GEMM: C(MxN) = A(MxK) @ B(KxN) ----------------
// One wave computes a 16x64 output tile using 4 accumulators.
// EPILOGUE: 0 = raw store, 1 = bias + relu.

template <int EPILOGUE>
__global__ void wmma_gemm_f32(const float* __restrict__ A, int lda,
                              const float* __restrict__ Bm, int ldb,
                              const float* __restrict__ bias,
                              float* __restrict__ C, int ldc,
                              int rowStrips, int totalWaves, int K) {
    int wave = (int)((blockIdx.x * (long)blockDim.x + threadIdx.x) >> 5);
    if (wave >= totalWaves) return;
    int lane = threadIdx.x & 31;
    int rs = wave % rowStrips;         // 16-row strip
    int cc = wave / rowStrips;         // 64-col chunk
    int row = rs * 16 + (lane & 15);
    int kh  = (lane >> 4) * 2;         // 0 (lanes 0-15) or 2 (lanes 16-31)
    int colBase = cc * 64 + (lane & 15);

    v8f acc[4] = {};
    for (int kk = 0; kk < K; kk += 4) {
        v2f a;
        a.x = A[row * lda + kk + kh];
        a.y = A[row * lda + kk + kh + 1];
#pragma unroll
        for (int t = 0; t < 4; ++t) {
            v2f b;
            b.x = Bm[(kk + kh) * ldb + colBase + t * 16];
            b.y = Bm[(kk + kh + 1) * ldb + colBase + t * 16];
            acc[t] = __builtin_amdgcn_wmma_f32_16x16x4_f32(
                false, a, false, b, (short)0, acc[t], false, false);
        }
    }

    int rOff = (lane < 16) ? 0 : 8;
#pragma unroll
    for (int t = 0; t < 4; ++t) {
        int col = colBase + t * 16;
        float bv = (EPILOGUE == 1) ? bias[col] : 0.0f;
#pragma unroll
        for (int r = 0; r < 8; ++r) {
            float v = acc[t][r];
            if (EPILOGUE == 1) v = fmaxf(v + bv, 0.0f);
            C[(long)(rs * 16 + r + rOff) * ldc + col] = v;
        }
    }
}

// ---------------- edge aggregation + combine ----------------

__global__ void agg64_kernel(const float* __restrict__ h, const int* __restrict__ ei,
                             const float* __restrict__ wedge, float* agg, int E) {
    int wid = (int)((blockIdx.x * (long)blockDim.x + threadIdx.x) >> 5);
    int lane = threadIdx.x & 31;
    if (wid >= E) return;
    int row = ei[wid];
    int col = ei[E + wid];
    float w = wedge[wid];
    unsafeAtomicAdd(&agg[(long)col * 64 + lane],      h[(long)row * 64 + lane]      * w);
    unsafeAtomicAdd(&agg[(long)col * 64 + lane + 32], h[(long)row * 64 + lane + 32] * w);
}

__global__ void agg1_kernel(const float* __restrict__ h, const int* __restrict__ ei,
                            const float* __restrict__ wedge, float* agg, int E) {
    int e = blockIdx.x * TPB + threadIdx.x;
    if (e >= E) return;
    unsafeAtomicAdd(&agg[ei[E + e]], h[ei[e]] * wedge[e]);
}

__global__ void combine64_kernel(const float* __restrict__ agg, const float* __restrict__ h,
                                 const float* __restrict__ dis, const float* __restrict__ b,
                                 float* out, int total) {
    int i = blockIdx.x * TPB + threadIdx.x;
    if (i >= total) return;
    int n = i >> 6;
    int f = i & 63;
    float dv = dis[n];
    out[i] = tanhf(agg[i] + dv * dv * h[i] + b[f]);
}

__global__ void combine1_kernel(const float* __restrict__ agg, const float* __restrict__ h,
                                const float* __restrict__ dis, const float* __restrict__ b,
                                float* out, int N) {
    int n = blockIdx.x * TPB + threadIdx.x;
    if (n >= N) return;
    float dv = dis[n];
    out[n] = tanhf(agg[n] + dv * dv * h[n] + b[0]);
}

__global__ void matvec64_kernel(const float* __restrict__ x, const float* __restrict__ W,
                                float* out, int N) {
    int n = blockIdx.x * TPB + threadIdx.x;
    if (n >= N) return;
    float s = 0.0f;
    const float* xr = x + (long)n * 64;
#pragma unroll
    for (int f = 0; f < 64; ++f) s += xr[f] * W[f];
    out[n] = s;
}

// ---------------- sort pooling ----------------

__global__ void sort_kernel(const float* __restrict__ x4, const int* __restrict__ nn,
                            const int* __restrict__ offs, int* sortidx) {
    int g = blockIdx.x, tid = threadIdx.x;
    __shared__ float key[1024];
    __shared__ int   idx[1024];
    int nNodes = nn[g], off = offs[g];
    for (int i = tid; i < 1024; i += TPB) {
        key[i] = (i < nNodes) ? x4[off + i] : -3.0e38f;
        idx[i] = i;
    }
    __syncthreads();
    for (int k = 2; k <= 1024; k <<= 1) {
        for (int j = k >> 1; j > 0; j >>= 1) {
            for (int i = tid; i < 1024; i += TPB) {
                int l = i ^ j;
                if (l > i) {
                    bool desc = ((i & k) == 0);
                    float ki = key[i], kl = key[l];
                    bool sw = desc ? (ki < kl) : (ki > kl);
                    if (sw) {
                        key[i] = kl; key[l] = ki;
                        int t = idx[i]; idx[i] = idx[l]; idx[l] = t;
                    }
                }
            }
            __syncthreads();
        }
    }
    for (int i = tid; i < 1024; i += TPB) sortidx[g * 1024 + i] = idx[i];
}

__device__ __forceinline__ float xcval(int d, long node,
                                       const float* x1, const float* x2,
                                       const float* x3, const float* x4) {
    if (d < 64)  return x1[node * 64 + d];
    if (d < 128) return x2[node * 64 + (d - 64)];
    if (d < 192) return x3[node * 64 + (d - 128)];
    return x4[node];
}

// sim / top-3 selection + pooled write; constants: D=193, K2=27, Nr=973, K1=3.
__global__ void simselect_kernel(const float* __restrict__ x1, const float* __restrict__ x2,
                                 const float* __restrict__ x3, const float* __restrict__ x4,
                                 const int* __restrict__ sortidx, const int* __restrict__ nn,
                                 const int* __restrict__ offs, const unsigned* __restrict__ minEnc,
                                 float* __restrict__ pooled) {
    const int D = 193, K2 = 27, Nr = 973;
    int g = blockIdx.x, tid = threadIdx.x;
    __shared__ float Srow[193];
    __shared__ float sq[973];
    __shared__ float sim[973];
    __shared__ float Tsum;
    __shared__ int   topIdx[3];
    int nNodes = nn[g], off = offs[g];
    int nv = min(nNodes, Nr);
    float fill = decf(*minEnc) - 1.0f;

    for (int d = tid; d < D; d += TPB) Srow[d] = 0.0f;
    if (tid == 0) Tsum = 0.0f;
    __syncthreads();

    float tlocal = 0.0f;
    for (int i = tid; i < nv; i += TPB) {
        int p = K2 + i;
        float s = 0.0f;
        if (p < nNodes) {
            long node = off + sortidx[g * 1024 + p];
            for (int d = 0; d < D; ++d) {
                float v = xcval(d, node, x1, x2, x3, x4);
                s += v * v;
                atomicAdd(&Srow[d], v);
            }
        } else {  // pad row: every feature == fill (reference quirk)
            s = (float)D * fill * fill;
            for (int d = 0; d < D; ++d) atomicAdd(&Srow[d], fill);
        }
        sq[i] = s;
        tlocal += s;
    }
    atomicAdd(&Tsum, tlocal);
    __syncthreads();

    float nf = (float)nv;
    for (int i = tid; i < nv; i += TPB) {
        int p = K2 + i;
        float dot = 0.0f;
        if (p < nNodes) {
            long node = off + sortidx[g * 1024 + p];
            for (int d = 0; d < D; ++d) dot += xcval(d, node, x1, x2, x3, x4) * Srow[d];
        } else {
            float ssum = 0.0f;
            for (int d = 0; d < D; ++d) ssum += Srow[d];
            dot = fill * ssum;
        }
        sim[i] = nf * sq[i] + Tsum - 2.0f * dot;
    }
    __syncthreads();

    if (tid == 0) {
        for (int t = 0; t < 3; ++t) {
            float best = -3.0e38f; int bi = 0;
            for (int i = 0; i < nv; ++i)
                if (sim[i] > best) { best = sim[i]; bi = i; }
            topIdx[t] = bi;
            sim[bi] = -3.4e38f;
        }
    }
    __syncthreads();

    for (int e = tid; e < 30 * D; e += TPB) {
        int r = e / D, d = e % D;
        float v;
        if (r < K2) {
            long node = off + sortidx[g * 1024 + r];   // graphs always have >= 27 nodes
            v = xcval(d, node, x1, x2, x3, x4);
        } else {
            int i = topIdx[r - K2];
            int p = K2 + i;
            if (p < nNodes) {
                long node = off + sortidx[g * 1024 + p];
                v = xcval(d, node, x1, x2, x3, x4);
            } else {
                v = fill;
            }
        }
        pooled[(long)g * 5790 + e] = v;
    }
}

// ---------------- conv tail ----------------

// conv5: per-graph (30 x 193) @ (193 x 64) + bias, relu; store channel-major (o,k).
__global__ void conv5_kernel(const float* __restrict__ pooled, const float* __restrict__ w5,
                             const float* __restrict__ b5, float* __restrict__ c5out) {
    int g = blockIdx.x, tid = threadIdx.x;
    __shared__ float P[5790];
    for (int i = tid; i < 5790; i += TPB) P[i] = pooled[(long)g * 5790 + i];
    __syncthreads();
    for (int e = tid; e < 30 * 64; e += TPB) {
        int k = e / 64, o = e % 64;
        float s = b5[o];
        const float* w = w5 + o * 193;
        const float* pr = P + k * 193;
        for (int i = 0; i < 193; ++i) s += pr[i] * w[i];
        c5out[(long)g * 1920 + o * 30 + k] = fmaxf(s, 0.0f);
    }
}

// maxpool pairs (30->15) then conv6 (64ch, k=5 -> len 11), relu; flatten (o*11+t).
__global__ void conv6_kernel(const float* __restrict__ c5out, const float* __restrict__ w6,
                             const float* __restrict__ b6, float* __restrict__ c6out) {
    int g = blockIdx.x, tid = threadIdx.x;
    __shared__ float P2[64 * 15];
    for (int e = tid; e < 64 * 15; e += TPB) {
        int c = e / 15, t = e % 15;
        float a = c5out[(long)g * 1920 + c * 30 + 2 * t];
        float b = c5out[(long)g * 1920 + c * 30 + 2 * t + 1];
        P2[e] = fmaxf(a, b);
    }
    __syncthreads();
    for (int e = tid; e < 64 * 11; e += TPB) {
        int o = e / 11, t = e % 11;
        float s = b6[o];
        const float* w = w6 + (long)o * 64 * 5;
        for (int c = 0; c < 64; ++c) {
#pragma unroll
            for (int j = 0; j < 5; ++j) s += P2[c * 15 + t + j] * w[c * 5 + j];
        }
        c6out[(long)g * 704 + o * 11 + t] = fmaxf(s, 0.0f);
    }
}

// ---------------- host orchestration ----------------

static inline int cdiv(long a, long b) { return (int)((a + b - 1) / b); }

extern "C" void kernel_launch(void* const* d_in, const int* in_sizes, int n_in,
                              void* d_out, int out_size, void* d_ws, size_t ws_size,
                              hipStream_t stream) {
    (void)n_in; (void)ws_size;
    const float* x     = (const float*)d_in[0];
    const int*   ei    = (const int*)d_in[1];
    const int*   batch = (const int*)d_in[2];
    const float* W1 = (const float*)d_in[3];  const float* b1 = (const float*)d_in[4];
    const float* W2 = (const float*)d_in[5];  const float* b2 = (const float*)d_in[6];
    const float* W3 = (const float*)d_in[7];  const float* b3 = (const float*)d_in[8];
    const float* W4 = (const float*)d_in[9];  const float* b4 = (const float*)d_in[10];
    const float* c5w = (const float*)d_in[11]; const float* c5b = (const float*)d_in[12];
    const float* c6w = (const float*)d_in[13]; const float* c6b = (const float*)d_in[14];
    const float* fcw = (const float*)d_in[15]; const float* fcb = (const float*)d_in[16];
    float* out = (float*)d_out;

    const int N = in_sizes[0] / 128;   // 230400
    const int E = in_sizes[1] / 2;     // 2000128
    const int B = out_size / 1600;     // 256

    // workspace carve (256B aligned)
    size_t off = 0;
    auto carve = [&](size_t bytes) {
        void* p = (char*)d_ws + off;
        off += (bytes + 255) & ~(size_t)255;
        return p;
    };
    float* dis     = (float*)carve((size_t)N * 4);
    float* wedge   = (float*)carve((size_t)E * 4);
    float* h       = (float*)carve((size_t)N * 64 * 4);   // also h4 (first N floats)
    float* agg     = (float*)carve((size_t)N * 64 * 4);   // also agg4
    float* x1      = (float*)carve((size_t)N * 64 * 4);
    float* x2      = (float*)carve((size_t)N * 64 * 4);
    float* x3      = (float*)carve((size_t)N * 64 * 4);
    float* x4      = (float*)carve((size_t)N * 4);
    int*   nn      = (int*)carve((size_t)B * 4);
    int*   offs    = (int*)carve((size_t)(B + 1) * 4);
    int*   sortidx = (int*)carve((size_t)B * 1024 * 4);
    unsigned* minEnc = (unsigned*)carve(256);
    float* pooled  = (float*)carve((size_t)B * 5790 * 4);
    float* c5out   = (float*)carve((size_t)B * 1920 * 4);
    float* c6out   = (float*)carve((size_t)B * 704 * 4);

    const long tot64 = (long)N * 64;

    // preprocessing
    zero_f32_kernel<<<2048, TPB, 0, stream>>>(dis, N);
    zero_f32_kernel<<<64, TPB, 0, stream>>>((float*)nn, B);
    deg_kernel<<<cdiv(E, TPB), TPB, 0, stream>>>(ei, dis, E);
    count_kernel<<<cdiv(N, TPB), TPB, 0, stream>>>(batch, nn, N);
    dis_kernel<<<cdiv(N, TPB), TPB, 0, stream>>>(dis, N);
    wedge_kernel<<<cdiv(E, TPB), TPB, 0, stream>>>(ei, dis, wedge, E);
    offsets_kernel<<<1, TPB, 0, stream>>>(nn, offs, B);

    // GCN layer helper (64-wide)
    auto gcn64 = [&](const float* xin, int K, const float* W, const float* bvec, float* xout) {
        int rowStrips = N / 16;
        int totalWaves = rowStrips;  // colChunks == 1
        wmma_gemm_f32<0><<<cdiv((long)totalWaves * 32, TPB), TPB, 0, stream>>>(
            xin, K, W, 64, nullptr, h, 64, rowStrips, totalWaves, K);
        zero_f32_kernel<<<4096, TPB, 0, stream>>>(agg, tot64);
        agg64_kernel<<<cdiv((long)E * 32, TPB), TPB, 0, stream>>>(h, ei, wedge, agg, E);
        combine64_kernel<<<cdiv(tot64, TPB), TPB, 0, stream>>>(agg, h, dis, bvec, xout, (int)tot64);
    };

    gcn64(x,  128, W1, b1, x1);
    gcn64(x1,  64, W2, b2, x2);
    gcn64(x2,  64, W3, b3, x3);

    // layer 4 (1-wide)
    matvec64_kernel<<<cdiv(N, TPB), TPB, 0, stream>>>(x3, W4, h, N);
    zero_f32_kernel<<<2048, TPB, 0, stream>>>(agg, N);
    agg1_kernel<<<cdiv(E, TPB), TPB, 0, stream>>>(h, ei, wedge, agg, E);
    combine1_kernel<<<cdiv(N, TPB), TPB, 0, stream>>>(agg, h, dis, b4, x4, N);

    // global min of xc (for fill value)
    min_init_kernel<<<1, 1, 0, stream>>>(minEnc);
    min_enc_kernel<<<1024, TPB, 0, stream>>>(x1, tot64, minEnc);
    min_enc_kernel<<<1024, TPB, 0, stream>>>(x2, tot64, minEnc);
    min_enc_kernel<<<1024, TPB, 0, stream>>>(x3, tot64, minEnc);
    min_enc_kernel<<<1024, TPB, 0, stream>>>(x4, N, minEnc);

    // sort pooling + selection
    sort_kernel<<<B, TPB, 0, stream>>>(x4, nn, offs, sortidx);
    simselect_kernel<<<B, TPB, 0, stream>>>(x1, x2, x3, x4, sortidx, nn, offs, minEnc, pooled);

    // conv tail
    conv5_kernel<<<B, TPB, 0, stream>>>(pooled, c5w, c5b, c5out);
    conv6_kernel<<<B, TPB, 0, stream>>>(c5out, c6w, c6b, c6out);

    // final FC: (B x 704) @ (704 x 1600) + bias, relu  -> out
    {
        int rowStrips = B / 16;                 // 16
        int colChunks = 1600 / 64;              // 25
        int totalWaves = rowStrips * colChunks; // 400
        wmma_gemm_f32<1><<<cdiv((long)totalWaves * 32, TPB), TPB, 0, stream>>>(
            c6out, 704, fcw, 1600, fcb, out, 1600, rowStrips, totalWaves, 704);
    }
}